// DynamicConv2d_17222818857085
// MI455X (gfx1250) — compile-verified
//
#include <hip/hip_runtime.h>
#include <stdint.h>

#define BDIM 256
constexpr int Bn = 64, Cc = 8, Hh = 256, Ww = 256, OC = 8, IF = 8;
constexpr int KK = Cc * 3 * 3;   // 72 reduction length
constexpr int TH = 16, TW = 32;  // output tile per pipeline step
constexpr int PH = TH + 2;       // 18 padded rows (halo)
constexpr int PW = TW + 4;       // 36 cols: 34 used (halo) + 2 pad -> TDM pitch
constexpr int NSTEP = KK / 4;    // 18 WMMA k-steps
constexpr int NTILE = Cc * PH * PW;  // 5184 floats per tile buffer
constexpr int WT = Ww / TW;      // 8 w-tiles per row strip

typedef float v2f __attribute__((ext_vector_type(2)));
typedef float v8f __attribute__((ext_vector_type(8)));
typedef unsigned int v4u __attribute__((ext_vector_type(4)));
typedef int v4i __attribute__((ext_vector_type(4)));
typedef int v8i __attribute__((ext_vector_type(8)));

// LDS element offset of im2col row kf = c*9 + kh*3 + kw (compile-time folded)
__device__ constexpr int xoff(int kf) {
  const int c = kf / 9, rem = kf % 9, kh = rem / 3, kw = rem % 3;
  return c * PH * PW + kh * PW + kw;
}

__device__ __forceinline__ uint32_t lds_off32(const void* p) {
  // addrspace(3) generic pointers carry the LDS offset in the low 32 bits
  return (uint32_t)(uintptr_t)p;
}

// One TDM descriptor: DMA a 2D tile (nrows x TW fp32, row stride W) from global
// into LDS at pitch PW, using pad_enable (interval 32 DW -> +4 DW padding).
__device__ __forceinline__ void tdm_load_channel(const float* gsrc,
                                                 uint32_t lds_addr, int nrows) {
  const uint64_t ga = (uint64_t)(uintptr_t)gsrc;
  const uint32_t ga_lo = (uint32_t)__builtin_amdgcn_readfirstlane((int)(uint32_t)ga);
  const uint32_t ga_hi = (uint32_t)__builtin_amdgcn_readfirstlane((int)(uint32_t)(ga >> 32));
  const uint32_t la    = (uint32_t)__builtin_amdgcn_readfirstlane((int)lds_addr);
  const int      nr    = __builtin_amdgcn_readfirstlane(nrows);

  v4u g0;                                   // D# group 0
  g0.x = 1u;                                // count=1, user descriptor
  g0.y = la;                                // lds_addr (bytes)
  g0.z = ga_lo;                             // global_addr[31:0]
  g0.w = (ga_hi & 0x01FFFFFFu) | (2u << 30);  // global_addr[56:32] | type=2

  v8i g1;                                   // D# group 1
  g1[0] = (int)((2u << 16)      // data_size = 4B
              | (1u << 20)      // pad_enable
              | (4u << 22)      // pad_interval: 2^(4+1)=32 DWORDs per row
              | (3u << 25));    // pad_amount: 3+1 = 4 DWORDs  -> pitch 36
  g1[1] = 0;                    // abar addr=0; tensor_dim0[15:0]=0
  g1[2] = 0x4000;               // tensor_dim0=0x4000_0000 (large, never OOB)
  g1[3] = (int)(0x4000u | ((uint32_t)TW << 16));  // tensor_dim1 large | tile_dim0=32
  g1[4] = nr;                   // tile_dim1 = nrows; tile_dim2 = 0 (2D)
  g1[5] = Ww;                   // tensor_dim0_stride = 256
  g1[6] = 0;
  g1[7] = 1;                    // tensor_dim1_stride = 65536 (ignored for 2D)

  v4i g2; g2[0] = 0x40000000; g2[1] = 0; g2[2] = 0; g2[3] = 0;
  v4i g3; g3[0] = 0; g3[1] = 0; g3[2] = 0; g3[3] = 0;

#if __clang_major__ >= 23
  v8i gx; gx[0]=0; gx[1]=0; gx[2]=0; gx[3]=0; gx[4]=0; gx[5]=0; gx[6]=0; gx[7]=0;
  __builtin_amdgcn_tensor_load_to_lds(g0, g1, g2, g3, gx, 0);
#else
  __builtin_amdgcn_tensor_load_to_lds(g0, g1, g2, g3, 0);
#endif
}

// Stage one zero-haloed x tile into LDS: wave i TDM-DMAs channel i's interior;
// halo/pad columns via async global->LDS b32 (or LDS zero stores when OOB).
__device__ __forceinline__ void stage_tile(const float* __restrict__ xb,
                                           float* dst, int h0, int w0,
                                           int tid, int wv) {
  const int rstart = (h0 == 0) ? 1 : 0;              // clip image top
  const int rend   = (h0 + TH == Hh) ? PH - 1 : PH;  // clip image bottom
  const int nrows  = rend - rstart;

  // interior (padded cols 1..TW): one TDM op per wave/channel
  {
    const int ch = __builtin_amdgcn_readfirstlane(wv);
    const uint32_t la = lds_off32(dst) +
                        (uint32_t)(((ch * PH + rstart) * PW + 1) * 4);
    const float* gsrc = xb + ((size_t)ch * Hh + (h0 - 1 + rstart)) * Ww + w0;
    tdm_load_channel(gsrc, la, nrows);
  }

  // halo cols {0, TW+1} + pad cols {TW+2, TW+3}
  for (int idx = tid; idx < Cc * PH * 4; idx += BDIM) {
    const int c  = idx / (PH * 4);
    const int rr = (idx / 4) % PH;
    const int cs = idx & 3;
    const int col = (cs == 0) ? 0 : (TW + cs);       // 0, 33, 34, 35
    const int gr = h0 - 1 + rr, gc = w0 - 1 + col;
    float* p = dst + (c * PH + rr) * PW + col;
    if (col < TW + 2 && (unsigned)gr < (unsigned)Hh && (unsigned)gc < (unsigned)Ww) {
      const uint32_t lds = lds_off32(p);
      const uint64_t ga  = (uint64_t)(uintptr_t)(xb + (c * Hh + gr) * Ww + gc);
      asm volatile("global_load_async_to_lds_b32 %0, %1, off"
                   :: "v"(lds), "v"(ga) : "memory");
    } else {
      *p = 0.f;
    }
  }

  // clipped top/bottom interior rows -> zeros
  if (rstart == 1)
    for (int idx = tid; idx < Cc * TW; idx += BDIM)
      dst[(idx / TW) * PH * PW + 1 + (idx % TW)] = 0.f;
  if (rend == PH - 1)
    for (int idx = tid; idx < Cc * TW; idx += BDIM)
      dst[(idx / TW) * PH * PW + (PH - 1) * PW + 1 + (idx % TW)] = 0.f;
}

__device__ __forceinline__ void wait_stage_and_barrier() {
  __builtin_amdgcn_s_wait_tensorcnt(0);              // TDM DMA retired
  asm volatile("s_wait_asynccnt 0x0" ::: "memory");  // halo async loads retired
  __syncthreads();                                   // cross-wave visibility
}

__global__ __launch_bounds__(BDIM) void dynconv_wmma(
    const float* __restrict__ x, const float* __restrict__ dw,
    const float* __restrict__ Wg, const float* __restrict__ bg,
    float* __restrict__ out) {
  __shared__ float Alds[KK * 16];      // [kflat][m], m=8..15 zero-padded
  __shared__ float xlds[2][NTILE];     // double-buffered zero-haloed x tiles

  const int tid = threadIdx.x;
  const int bid = blockIdx.x;
  const int b  = bid / (Hh / TH);      // one (batch, row-strip) per block
  const int h0 = (bid % (Hh / TH)) * TH;

  // ---- phase 0: dynamic kernel generation into LDS (A matrix, 72x16) ----
  // view-reshape aliasing: kernels[b,oc,c,kh,kw] = dw[b,oc,:]·Wg[kflat,:] + bg[kflat]
  for (int idx = tid; idx < KK * 16; idx += BDIM) {
    const int m = idx & 15, kf = idx >> 4;
    float v = 0.f;
    if (m < OC) {
      v = bg[kf];
      const float* dwp = dw + (b * Cc + m) * IF;
      const float* wgp = Wg + kf * IF;
#pragma unroll
      for (int i = 0; i < IF; ++i) v = fmaf(dwp[i], wgp[i], v);
    }
    Alds[idx] = v;
  }

  const int lane = tid & 31, wv = tid >> 5;
  const int half = lane >> 4;      // K-pair select per 16x4 A / 4x16 B layout
  const int m    = lane & 15;      // M row for A, N pixel for B/D

  // ---- prologue: stage first tile (TDM + async halo) ----
  const float* xb = x + (size_t)b * Cc * Hh * Ww;
  stage_tile(xb, xlds[0], h0, 0, tid, wv);
  wait_stage_and_barrier();

  // per-lane B-fragment LDS element offsets (constants selected by lane half)
  int off0[NSTEP], off1[NSTEP];
#pragma unroll
  for (int s = 0; s < NSTEP; ++s) {
    off0[s] = half ? xoff(4 * s + 2) : xoff(4 * s);
    off1[s] = half ? xoff(4 * s + 3) : xoff(4 * s + 1);
  }

  // A fragments (reused for all 8 w-tiles): VGPR0=K{0|2}, VGPR1=K{1|3}
  float a0[NSTEP], a1[NSTEP];
#pragma unroll
  for (int s = 0; s < NSTEP; ++s) {
    const int k0 = 4 * s + 2 * half;
    a0[s] = Alds[k0 * 16 + m];
    a1[s] = Alds[(k0 + 1) * 16 + m];
  }

  // ---- pipelined loop over w-tiles: TDM-load next while WMMA on current ----
  for (int wt = 0; wt < WT; ++wt) {
    const int cur = wt & 1;
    if (wt + 1 < WT)
      stage_tile(xb, xlds[cur ^ 1], h0, (wt + 1) * TW, tid, wv);

    const float* xl = xlds[cur];
    const int w0 = wt * TW;

    for (int rt = 0; rt < 2; ++rt) {
      const int r = wv + rt * 8;                 // output row within tile
      for (int g = 0; g < TW / 16; ++g) {
        const int base = r * PW + g * 16 + m;    // padded-coords pixel base
        v8f acc = {0.f, 0.f, 0.f, 0.f, 0.f, 0.f, 0.f, 0.f};
#pragma unroll
        for (int s = 0; s < NSTEP; ++s) {
          v2f av, bv;
          av.x = a0[s];            av.y = a1[s];
          bv.x = xl[base + off0[s]];
          bv.y = xl[base + off1[s]];
          acc = __builtin_amdgcn_wmma_f32_16x16x4_f32(
              false, av, false, bv, (short)0, acc, false, false);
        }
        // D: VGPR v -> M=v (lanes 0-15) / M=v+8 (lanes 16-31, padding -> drop)
        if (lane < 16) {
          const int oh = h0 + r, ow = w0 + g * 16 + m;
#pragma unroll
          for (int v = 0; v < OC; ++v)
            out[(((size_t)b * OC + v) * Hh + oh) * Ww + ow] = acc[v];
        }
      }
    }

    wait_stage_and_barrier();   // retire next-tile DMA + async, then barrier
  }
}

extern "C" void kernel_launch(void* const* d_in, const int* in_sizes, int n_in,
                              void* d_out, int out_size, void* d_ws, size_t ws_size,
                              hipStream_t stream) {
  const float* x  = (const float*)d_in[0];   // (64, 8, 256, 256)
  const float* dw = (const float*)d_in[1];   // (64, 8, 8)
  const float* Wg = (const float*)d_in[2];   // (72, 8)
  const float* bg = (const float*)d_in[3];   // (72,)
  float* out = (float*)d_out;                // (64, 8, 256, 256)

  const int grid = Bn * (Hh / TH);           // 1024 persistent row-strip blocks
  dynconv_wmma<<<grid, BDIM, 0, stream>>>(x, dw, Wg, bg, out);
}